// SemiConv2d_72988674228358
// MI455X (gfx1250) — compile-verified
//
#include <hip/hip_runtime.h>
#include <cstdint>
#include <cstddef>

#define N_ 8
#define C_ 32
#define H_ 96
#define W_ 96
#define SXW 104   // padded x-row stride: [3] = col -1 (-inf), [4..99] = cols 0..95, [100] = col 96 (-inf)

// Fused semiring term: D = max(min(a,b), c)  ->  single v_maxmin_num_f32.
__device__ __forceinline__ float maxmin_f32(float a, float b, float c) {
    float d;
    asm("v_maxmin_num_f32 %0, %1, %2, %3" : "=v"(d) : "v"(a), "v"(b), "v"(c));
    return d;
}

// Block: 192 threads (6 waves, wave32). Each block: one batch n, a 2-row
// output strip, all 96 cols, all 32 oc.
// Thread: ocg = tid/24 -> oc0 = 4*ocg ; colg = tid%24 -> w0 = 4*colg.
// acc[2 rows][4 cols][4 oc] = 32 VGPRs.
//
// LDS: sK [(ic*9+tap)*32+oc] = 36 KB (transposed, oc innermost, loaded once)
//      sX [(r*16+icl)*104 + 4+col] = 26 KB, async-staged per 16-ic chunk via
//      global_load_async_to_lds_b128; border cells pre-filled with -inf.

__global__ __launch_bounds__(192) void semiconv2d_maxmin_kernel(
    const float* __restrict__ x, const float* __restrict__ kern,
    float* __restrict__ out)
{
    __shared__ float sK[C_ * 9 * C_];
    __shared__ float sX[4 * 16 * SXW];

    const int tid = threadIdx.x;
    const int nb  = blockIdx.x;                 // 0..383
    const int n   = nb / (H_ / 2);
    const int h0  = (nb % (H_ / 2)) * 2;

    const float NEG_INF = -__builtin_inff();

    // ---- Stage K transposed: global (oc,ic,kh,kw) -> sK[(ic*9+tap)*32+oc].
    for (int e = tid; e < C_ * C_ * 9; e += 192) {
        int oc  = e / (C_ * 9);
        int rem = e - oc * (C_ * 9);            // ic*9 + tap
        sK[rem * C_ + oc] = kern[e];
    }

    // ---- Fill the -inf border cells once (async loads never overwrite them).
    for (int e = tid; e < 4 * 16 * 2; e += 192) {
        int ri   = e >> 1;                       // (r*16+icl)
        int side = e & 1;
        sX[ri * SXW + (side ? 100 : 3)] = NEG_INF;
    }

    const int ocg  = tid / 24;
    const int colg = tid - ocg * 24;
    const int oc0  = ocg * 4;
    const int w0   = colg * 4;

    float acc[2][4][4];
#pragma unroll
    for (int rr = 0; rr < 2; ++rr)
#pragma unroll
        for (int p = 0; p < 4; ++p)
#pragma unroll
            for (int q = 0; q < 4; ++q) acc[rr][p][q] = NEG_INF;

    for (int c = 0; c < 2; ++c) {
        const int icb = c * 16;

        // ---- Async-stage x rows h0-1 .. h0+2 for 16 input channels.
        // 4 rows * 16 ic * 24 col-quads = 1536 slots / 192 threads = 8 each.
#pragma unroll
        for (int j = 0; j < 8; ++j) {
            int s    = tid + 192 * j;
            int r    = s / 384;
            int rem  = s - r * 384;
            int icl  = rem / 24;
            int col  = (rem - icl * 24) * 4;
            int row  = h0 - 1 + r;
            float* ldst = &sX[(r * 16 + icl) * SXW + 4 + col];   // 16B aligned
            if (row >= 0 && row < H_) {
                const float* gsrc =
                    x + (((size_t)(n * C_ + icb + icl) * H_ + row) * W_ + col);
                uint32_t la = (uint32_t)(uintptr_t)ldst;
                uint64_t ga = (uint64_t)(uintptr_t)gsrc;
                asm volatile("global_load_async_to_lds_b128 %0, %1, off"
                             :: "v"(la), "v"(ga) : "memory");
            } else {
                *(float4*)ldst = make_float4(NEG_INF, NEG_INF, NEG_INF, NEG_INF);
            }
        }

#if defined(__has_builtin) && __has_builtin(__builtin_amdgcn_s_wait_asynccnt)
        __builtin_amdgcn_s_wait_asynccnt(0);
#else
        asm volatile("s_wait_asynccnt 0" ::: "memory");
#endif
        __syncthreads();

        // ---- Compute over this 16-ic chunk.
        for (int icl = 0; icl < 16; ++icl) {
            // 9 taps x 4 oc of K: 9 aligned ds_load_b128 (broadcast per ocg).
            float4 kv[9];
#pragma unroll
            for (int t = 0; t < 9; ++t)
                kv[t] = *(const float4*)&sK[((icb + icl) * 9 + t) * C_ + oc0];

            // 4 rows x 6 cols of x; borders come from the -inf pad cells.
            float xv[4][6];
#pragma unroll
            for (int r = 0; r < 4; ++r)
#pragma unroll
                for (int j = 0; j < 6; ++j)
                    xv[r][j] = sX[(r * 16 + icl) * SXW + 3 + w0 + j];

            // 2 rows * 4 cols * 9 taps * 4 oc = 288 v_maxmin_num_f32.
#pragma unroll
            for (int rr = 0; rr < 2; ++rr)
#pragma unroll
                for (int p = 0; p < 4; ++p)
#pragma unroll
                    for (int kh = 0; kh < 3; ++kh)
#pragma unroll
                        for (int kw = 0; kw < 3; ++kw) {
                            float xval = xv[rr + kh][p + kw];
                            float4 kk  = kv[kh * 3 + kw];
                            acc[rr][p][0] = maxmin_f32(xval, kk.x, acc[rr][p][0]);
                            acc[rr][p][1] = maxmin_f32(xval, kk.y, acc[rr][p][1]);
                            acc[rr][p][2] = maxmin_f32(xval, kk.z, acc[rr][p][2]);
                            acc[rr][p][3] = maxmin_f32(xval, kk.w, acc[rr][p][3]);
                        }
        }
        if (c == 0) __syncthreads();   // all waves done reading before restage
    }

    // ---- Write 2 rows x 4 oc x 4 cols as aligned b128 stores.
#pragma unroll
    for (int rr = 0; rr < 2; ++rr)
#pragma unroll
        for (int q = 0; q < 4; ++q) {
            float4 v = make_float4(acc[rr][0][q], acc[rr][1][q],
                                   acc[rr][2][q], acc[rr][3][q]);
            *(float4*)&out[((size_t)(n * C_ + oc0 + q) * H_ + (h0 + rr)) * W_ + w0] = v;
        }
}

extern "C" void kernel_launch(void* const* d_in, const int* in_sizes, int n_in,
                              void* d_out, int out_size, void* d_ws, size_t ws_size,
                              hipStream_t stream) {
    (void)in_sizes; (void)n_in; (void)out_size; (void)d_ws; (void)ws_size;
    const float* x    = (const float*)d_in[0];
    const float* kern = (const float*)d_in[1];
    float*       out  = (float*)d_out;

    dim3 grid(N_ * (H_ / 2));   // 384 blocks: (n, 2-row strip)
    dim3 block(192);            // 6 waves of 32
    hipLaunchKernelGGL(semiconv2d_maxmin_kernel, grid, block, 0, stream,
                       x, kern, out);
}